// PathRewardWithCollision_20298015441021
// MI455X (gfx1250) — compile-verified
//
#include <hip/hip_runtime.h>
#include <math.h>

#define A_ 128
#define M_ 64
#define N_ 64
#define T_ 80

typedef __attribute__((ext_vector_type(2))) float v2f;
typedef __attribute__((ext_vector_type(8))) float v8f;

// ---------------------------------------------------------------------------
// Kernel 1: l2_pen[a,m] = -(sum_t ||(tp - tg) * mask|| - 8)
// ---------------------------------------------------------------------------
__global__ void l2_kernel(const float* __restrict__ targets,
                          const float* __restrict__ target_mask,
                          const float* __restrict__ traj_preds,
                          float* __restrict__ l2_pen_ws,
                          float* __restrict__ out) {
  int idx = blockIdx.x * blockDim.x + threadIdx.x;   // idx = a*M + m
  if (idx >= A_ * M_) return;
  int a = idx / M_;
  const float4* tp = (const float4*)(traj_preds + (size_t)idx * T_ * 2);
  const float4* tg = (const float4*)(targets    + (size_t)a   * T_ * 2);
  const float*  tm = target_mask + (size_t)a * T_;
  float s = 0.f;
  for (int t2 = 0; t2 < T_ / 2; ++t2) {              // 2 timesteps per float4
    float4 p = tp[t2];
    float4 g = tg[t2];
    float m0 = tm[2 * t2], m1 = tm[2 * t2 + 1];
    float dx0 = (p.x - g.x) * m0, dy0 = (p.y - g.y) * m0;
    float dx1 = (p.z - g.z) * m1, dy1 = (p.w - g.w) * m1;
    s += sqrtf(dx0 * dx0 + dy0 * dy0) + sqrtf(dx1 * dx1 + dy1 * dy1);
  }
  float pen = -(s - 8.0f);
  l2_pen_ws[idx] = pen;
  out[2 * A_ * M_ + idx] = pen;                      // l2_pen output
}

// ---------------------------------------------------------------------------
// Kernel 2: collision grid via V_WMMA_F32_16X16X4_F32.
//   A-row(m) = ( x, y, |p|^2, 1 ),  B-col(n) = ( -2x, -2y, 1, |q|^2 )
//   => C[m,n] = dist^2(m,n) in one matrix op; epilogue is one compare.
//
// Validity folding: when target_mask[t]==0 the hit must be false regardless
// of distance, so (a) p needs no masking here, and (b) the staged half-width
// is set to NaN -> thr2 becomes NaN -> (dist2 < NaN) == false. Branchless.
//
// grid = A blocks of 128 threads; wave w owns m-tile w; loops nt(4) x t(80),
// t chunked by 4 so thresholds come from one ds_load_b128 per 4 WMMAs.
// ---------------------------------------------------------------------------
__global__ void __launch_bounds__(128)
coll_kernel(const float* __restrict__ target_mask,
            const float* __restrict__ traj_preds,
            const float* __restrict__ agent_fut_width,
            const float* __restrict__ other_trajs,
            const unsigned char* __restrict__ other_masks,
            const float* __restrict__ other_widths,
            const float* __restrict__ l2_pen_ws,
            float* __restrict__ out) {
  __shared__ __align__(16) float s_hwv[T_];  // 0.5*width, or NaN if masked out

  const int a    = blockIdx.x;
  const int wave = threadIdx.x >> 5;      // m-tile index 0..3
  const int lane = threadIdx.x & 31;
  const int half = lane >> 4;             // 0: K0/K1 lanes, 1: K2/K3 lanes
  const int lm   = lane & 15;

  for (int t = threadIdx.x; t < T_; t += 128) {
    float tm = target_mask[a * T_ + t];
    float hw = 0.5f * agent_fut_width[a * T_ + t];
    s_hwv[t] = (tm != 0.f) ? hw : __builtin_nanf("");
  }
  __syncthreads();                        // reconverged: EXEC all-ones below

  const int m = wave * 16 + lm;           // row this lane loads (both halves)
  const float4* tp = (const float4*)(traj_preds + (size_t)(a * M_ + m) * T_ * 2);

  float count[8];
#pragma unroll
  for (int j = 0; j < 8; ++j) count[j] = 0.f;

  for (int nt = 0; nt < N_ / 16; ++nt) {
    const int n = nt * 16 + lm;           // column this lane loads
    const float4* q4 =
        (const float4*)(other_trajs + (size_t)(a * N_ + n) * T_ * 2);
    const unsigned int* mo4 =
        (const unsigned int*)(other_masks + (size_t)(a * N_ + n) * T_);
    const float wo_h = 0.5f * other_widths[a * N_ + n];

    float hit[8];
#pragma unroll
    for (int j = 0; j < 8; ++j) hit[j] = 0.f;

    for (int tq = 0; tq < T_ / 4; ++tq) { // 4 timesteps per iteration
      float4 pA = tp[2 * tq], pB = tp[2 * tq + 1];
      float4 qA = q4[2 * tq], qB = q4[2 * tq + 1];
      unsigned int mq = mo4[tq];          // 4 mask bytes
      float4 hw4 = *(const float4*)&s_hwv[4 * tq];

      float px4[4] = {pA.x, pA.z, pB.x, pB.z};
      float py4[4] = {pA.y, pA.w, pB.y, pB.w};
      float qx4[4] = {qA.x, qA.z, qB.x, qB.z};
      float qy4[4] = {qA.y, qA.w, qB.y, qB.w};
      float hwa[4] = {hw4.x, hw4.y, hw4.z, hw4.w};

#pragma unroll
      for (int u = 0; u < 4; ++u) {
        float px = px4[u], py = py4[u];
        float qx = qx4[u], qy = qy4[u];

        v2f av, bv;
        // A 16x4 f32 layout: lanes 0-15 = K0,K1 ; lanes 16-31 = K2,K3
        av.x = half ? (px * px + py * py) : px;
        av.y = half ? 1.0f                : py;
        // B 4x16 f32 layout mirrors A
        bv.x = half ? 1.0f                : (-2.0f * qx);
        bv.y = half ? (qx * qx + qy * qy) : (-2.0f * qy);

        v8f c = {};
        c = __builtin_amdgcn_wmma_f32_16x16x4_f32(
            false, av, false, bv, (short)0, c, false, false);

        float comb = hwa[u] + wo_h;       // NaN if target_mask[t]==0
        float thr2 = comb * comb;         // NaN propagates
        bool  mvalid = ((mq >> (8 * u)) & 0xFF) != 0;
        float tv = mvalid ? thr2 : -1.0f; // dist^2 never < -1; NaN cmp false
#pragma unroll
        for (int j = 0; j < 8; ++j) hit[j] = (c[j] < tv) ? 1.f : hit[j];
      }
    }
#pragma unroll
    for (int j = 0; j < 8; ++j) count[j] += hit[j];
  }

  // Sum over the 16 n-lanes within each half (masks 8,4,2,1 stay in-half).
#pragma unroll
  for (int off = 8; off >= 1; off >>= 1) {
#pragma unroll
    for (int j = 0; j < 8; ++j) count[j] += __shfl_xor(count[j], off, 32);
  }

  if (lm == 0) {
    // C/D layout: vgpr j holds M = half*8 + j for this half.
#pragma unroll
    for (int j = 0; j < 8; ++j) {
      int idx = a * M_ + wave * 16 + half * 8 + j;
      float coll_pen = -count[j] * 100.f;
      float l2p = l2_pen_ws[idx];
      out[idx] = l2p + coll_pen;          // total
      out[A_ * M_ + idx] = coll_pen;      // coll_pen
    }
  }
}

// ---------------------------------------------------------------------------
extern "C" void kernel_launch(void* const* d_in, const int* in_sizes, int n_in,
                              void* d_out, int out_size, void* d_ws, size_t ws_size,
                              hipStream_t stream) {
  const float*         targets          = (const float*)d_in[0];
  const float*         target_mask      = (const float*)d_in[1];
  const float*         traj_preds       = (const float*)d_in[2];
  /* d_in[3] traj_scores: unused by the reference */
  const float*         agent_fut_width  = (const float*)d_in[4];
  const float*         other_fut_trajs  = (const float*)d_in[5];
  const unsigned char* other_fut_masks  = (const unsigned char*)d_in[6];
  const float*         other_fut_widths = (const float*)d_in[7];
  float* out = (float*)d_out;
  float* ws  = (float*)d_ws;   // 8192 floats: l2_pen scratch

  l2_kernel<<<(A_ * M_ + 255) / 256, 256, 0, stream>>>(
      targets, target_mask, traj_preds, ws, out);

  coll_kernel<<<A_, 128, 0, stream>>>(
      target_mask, traj_preds, agent_fut_width,
      other_fut_trajs, other_fut_masks, other_fut_widths, ws, out);
}